// DataProcessing_4105988735041
// MI455X (gfx1250) — compile-verified
//
#include <hip/hip_runtime.h>

// ---------------- problem constants ----------------
#define TLEN     4800000       // samples
#define HOP      128
#define NFFT     512
#define NB       257           // n_freqs == n_mels
#define NBP      272           // 257 padded to multiple of 16
#define F_FRAMES 37501         // 1 + TLEN/HOP
#define FP       37504         // frames padded to multiple of 16
#define CH       4

typedef float v2f __attribute__((ext_vector_type(2)));
typedef float v8f __attribute__((ext_vector_type(8)));

// ---------------- frame extraction (reflect pad, rect + hann) ----------------
__global__ void build_frames(const float* __restrict__ sig,
                             float* __restrict__ fr, float* __restrict__ fh) {
  long gid = (long)blockIdx.x * blockDim.x + threadIdx.x;
  if (gid >= (long)FP * NFFT) return;
  int f = (int)(gid / NFFT);
  int n = (int)(gid % NFFT);
  float x = 0.0f;
  if (f < F_FRAMES) {
    long j = (long)f * HOP + n - (NFFT / 2);
    if (j < 0) j = -j;                       // reflect left
    if (j >= TLEN) j = 2L * TLEN - 2 - j;    // reflect right
    x = sig[j];
  }
  float h = 0.5f * (1.0f - cosf(6.2831853071795864f * (float)n / (float)NFFT));
  fr[gid] = x;
  fh[gid] = x * h;
}

// ---------------- DFT basis: cos / -sin, (NFFT x NBP), cols >=257 zero ------
__global__ void build_basis(float* __restrict__ cb, float* __restrict__ sb) {
  int gid = blockIdx.x * blockDim.x + threadIdx.x;
  if (gid >= NFFT * NBP) return;
  int n = gid / NBP;
  int k = gid % NBP;
  float c = 0.0f, s = 0.0f;
  if (k < NB) {
    int t = (n * k) & (NFFT - 1);            // modular reduction keeps angle small
    float ang = (6.2831853071795864f / (float)NFFT) * (float)t;
    c = cosf(ang);
    s = -sinf(ang);
  }
  cb[gid] = c;
  sb[gid] = s;
}

// ---------------- mel filterbank zero-padded to 272x272 ---------------------
__global__ void build_fbpad(const float* __restrict__ fb, float* __restrict__ fbp) {
  int gid = blockIdx.x * blockDim.x + threadIdx.x;
  if (gid >= NBP * NBP) return;
  int r = gid / NBP, c = gid % NBP;
  fbp[gid] = (r < NB && c < NB) ? fb[r * NB + c] : 0.0f;
}

// ---------------- WMMA DFT: spec = frames(FPx512) x basis(512xNBP) ----------
// blockIdx.x selects one 16-column slab of the basis, shared by all 8 waves:
// stage cos/-sin slabs (512x16 each, 64 KB total) in LDS once per block,
// then each wave computes a 16x16 tile with dual Re/Im accumulators.
__global__ void wmma_dft(const float* __restrict__ A,
                         const float* __restrict__ CB,
                         const float* __restrict__ SB,
                         float* __restrict__ Re, float* __restrict__ Im) {
  __shared__ float cs[NFFT * 16];
  __shared__ float ss[NFFT * 16];
  int tid  = threadIdx.x;
  int tn   = blockIdx.x;                     // 0..16 (block-uniform)
  int base = tn * 16;
  for (int e = tid; e < NFFT * 16; e += 256) {
    int k = e >> 4, n = e & 15;
    cs[e] = CB[k * NBP + base + n];
    ss[e] = SB[k * NBP + base + n];
  }
  __syncthreads();

  int lane = tid & 31;
  int wave = tid >> 5;
  int tm = blockIdx.y * 8 + wave;            // wave-uniform -> EXEC all-1
  if (tm >= FP / 16) return;
  int half = lane >> 4, l = lane & 15;
  int m = tm * 16 + l;
  const float* arow = A + (long)m * NFFT + 2 * half;
  __builtin_prefetch(arow, 0, 1);            // global_prefetch_b8
  v8f cre = {0,0,0,0,0,0,0,0};
  v8f cim = {0,0,0,0,0,0,0,0};
  for (int k = 0; k < NFFT; k += 4) {
    v2f a;  a.x  = arow[k];                  a.y  = arow[k + 1];
    int kr = k + 2 * half;
    v2f br; br.x = cs[kr * 16 + l];          br.y = cs[kr * 16 + 16 + l];
    v2f bi; bi.x = ss[kr * 16 + l];          bi.y = ss[kr * 16 + 16 + l];
    cre = __builtin_amdgcn_wmma_f32_16x16x4_f32(false, a, false, br, (short)0, cre, false, false);
    cim = __builtin_amdgcn_wmma_f32_16x16x4_f32(false, a, false, bi, (short)0, cim, false, false);
  }
  for (int r = 0; r < 8; ++r) {
    long row = (long)(tm * 16 + r + 8 * half);
    Re[row * NBP + base + l] = cre[r];
    Im[row * NBP + base + l] = cim[r];
  }
}

// ---------------- hann power spectrum, padded cols zeroed -------------------
__global__ void power_spec(const float* __restrict__ re, const float* __restrict__ im,
                           float* __restrict__ p) {
  long gid = (long)blockIdx.x * blockDim.x + threadIdx.x;
  if (gid >= (long)FP * NBP) return;
  int c = (int)(gid % NBP);
  float r = re[gid], i = im[gid];
  p[gid] = (c < NB) ? (r * r + i * i) : 0.0f;
}

// ---------------- WMMA mel GEMM: mel = power(FPxNBP) x fbp(NBPxNBP) ---------
__global__ void wmma_mel(const float* __restrict__ A, const float* __restrict__ B,
                         float* __restrict__ C) {
  __shared__ float bs[NBP * 16];             // 17 KB filterbank slab
  int tid  = threadIdx.x;
  int tn   = blockIdx.x;
  int base = tn * 16;
  for (int e = tid; e < NBP * 16; e += 256) {
    int k = e >> 4, n = e & 15;
    bs[e] = B[k * NBP + base + n];
  }
  __syncthreads();

  int lane = tid & 31;
  int wave = tid >> 5;
  int tm = blockIdx.y * 8 + wave;
  if (tm >= FP / 16) return;
  int half = lane >> 4, l = lane & 15;
  int m = tm * 16 + l;
  const float* arow = A + (long)m * NBP + 2 * half;
  v8f acc = {0,0,0,0,0,0,0,0};
  for (int k = 0; k < NBP; k += 4) {
    v2f a; a.x = arow[k];                    a.y = arow[k + 1];
    int kr = k + 2 * half;
    v2f b; b.x = bs[kr * 16 + l];            b.y = bs[kr * 16 + 16 + l];
    acc = __builtin_amdgcn_wmma_f32_16x16x4_f32(false, a, false, b, (short)0, acc, false, false);
  }
  for (int r = 0; r < 8; ++r) {
    long row = (long)(tm * 16 + r + 8 * half);
    C[row * NBP + base + l] = acc[r];
  }
}

// ---------------- PCEN: sequential IIR per mel channel ----------------------
__global__ void pcen_kernel(const float* __restrict__ mel, float* __restrict__ out) {
  int c = blockIdx.x * blockDim.x + threadIdx.x;
  if (c >= NB) return;
  const float S = 0.025f, EPS = 1e-6f, ALPHA = 0.98f, DELTA = 2.0f;
  const float DR = 1.4142135623730951f;      // DELTA^0.5
  float m = 0.0f;
  for (int f = 0; f < F_FRAMES; ++f) {
    float x = mel[(long)f * NBP + c];
    m = (f == 0) ? (S * x) : ((1.0f - S) * m + S * x);
    float p = powf(m + EPS, ALPHA);
    out[((long)f * CH + 1) * NB + c] = sqrtf(x / p + DELTA) - DR;
  }
}

// ---------------- log-mag + demodulated phase (elementwise) -----------------
// unwrap adds only multiples of 2*pi => sin/cos(unwrapped) == im/|X|, re/|X|.
__global__ void assemble(const float* __restrict__ re, const float* __restrict__ im,
                         float* __restrict__ out) {
  long gid = (long)blockIdx.x * blockDim.x + threadIdx.x;
  if (gid >= (long)F_FRAMES * NB) return;
  long f = gid / NB;
  int  k = (int)(gid % NB);
  float r = re[f * NBP + k], i = im[f * NBP + k];
  float mag = sqrtf(r * r + i * i);
  out[(f * CH + 0) * NB + k] = logf(mag + 1e-9f);
  float inv = (mag > 0.0f) ? (1.0f / mag) : 0.0f;
  out[(f * CH + 2) * NB + k] = i * inv;                       // sin(angle)
  out[(f * CH + 3) * NB + k] = (mag > 0.0f) ? r * inv : 1.0f; // cos(angle)
}

// ---------------- L2 norm over frames, per (channel,freq) column ------------
__global__ void col_norm(const float* __restrict__ out, float* __restrict__ invn) {
  int j = blockIdx.x * blockDim.x + threadIdx.x;
  if (j >= CH * NB) return;
  float s = 0.0f;
  for (int f = 0; f < F_FRAMES; ++f) {
    float v = out[(long)f * CH * NB + j];
    s += v * v;
  }
  float nrm = sqrtf(s);
  invn[j] = 1.0f / fmaxf(nrm, 1e-12f);
}

__global__ void scale_out(float* __restrict__ out, const float* __restrict__ invn) {
  long gid = (long)blockIdx.x * blockDim.x + threadIdx.x;
  if (gid >= (long)F_FRAMES * CH * NB) return;
  out[gid] *= invn[gid % (CH * NB)];
}

// ---------------- launcher --------------------------------------------------
extern "C" void kernel_launch(void* const* d_in, const int* in_sizes, int n_in,
                              void* d_out, int out_size, void* d_ws, size_t ws_size,
                              hipStream_t stream) {
  const float* audio = (const float*)d_in[0];     // (1, TLEN) f32
  const float* fb    = (const float*)d_in[1];     // (257, 257) f32
  float* out = (float*)d_out;                     // (F, 4, 257) f32
  float* ws  = (float*)d_ws;

  size_t off = 0;
  float* cb      = ws + off; off += (size_t)NFFT * NBP;   // cos basis
  float* sb      = ws + off; off += (size_t)NFFT * NBP;   // -sin basis
  float* fbp     = ws + off; off += (size_t)NBP * NBP;    // padded filterbank
  float* frRect  = ws + off; off += (size_t)FP * NFFT;
  float* frHann  = ws + off; off += (size_t)FP * NFFT;
  float* specRe  = ws + off; off += (size_t)FP * NBP;
  float* specIm  = ws + off; off += (size_t)FP * NBP;
  float* hRe     = ws + off; off += (size_t)FP * NBP;
  float* hIm     = ws + off; off += (size_t)FP * NBP;
  float* melPow  = ws + off; off += (size_t)FP * NBP;
  float* mel     = ws + off; off += (size_t)FP * NBP;
  float* invn    = ws + off; off += (size_t)CH * NB;

  const int B = 256;
  long nFr = (long)FP * NFFT;
  build_frames<<<(unsigned)((nFr + B - 1) / B), B, 0, stream>>>(audio, frRect, frHann);
  build_basis <<<(NFFT * NBP + B - 1) / B, B, 0, stream>>>(cb, sb);
  build_fbpad <<<(NBP * NBP + B - 1) / B, B, 0, stream>>>(fb, fbp);

  dim3 gGemm(NBP / 16, (FP / 16 + 7) / 8);        // 17 x 293, 8 waves/block
  wmma_dft<<<gGemm, B, 0, stream>>>(frRect, cb, sb, specRe, specIm);
  wmma_dft<<<gGemm, B, 0, stream>>>(frHann, cb, sb, hRe, hIm);

  long nSp = (long)FP * NBP;
  power_spec<<<(unsigned)((nSp + B - 1) / B), B, 0, stream>>>(hRe, hIm, melPow);
  wmma_mel<<<gGemm, B, 0, stream>>>(melPow, fbp, mel);

  pcen_kernel<<<(NB + 63) / 64, 64, 0, stream>>>(mel, out);

  long nAs = (long)F_FRAMES * NB;
  assemble<<<(unsigned)((nAs + B - 1) / B), B, 0, stream>>>(specRe, specIm, out);

  col_norm<<<(CH * NB + B - 1) / B, B, 0, stream>>>(out, invn);
  long nOut = (long)F_FRAMES * CH * NB;
  scale_out<<<(unsigned)((nOut + B - 1) / B), B, 0, stream>>>(out, invn);
}